// MMDLinear_80315888435443
// MI455X (gfx1250) — compile-verified
//
#include <hip/hip_runtime.h>
#include <hip/hip_bf16.h>

// MMD-linear: mean(ΔΔᵀ) == ||colsum(Δ)||² / N².  N=8192, D=2048, f32.
// Stage 1: bandwidth-bound streaming column-sum (134 MB read once).
// Stage 2: cross-block partial reduction via V_WMMA_F32_16X16X4_F32 (1ᵀ·P).
// Stage 3: final 32-way reduce + scale.

typedef __attribute__((ext_vector_type(2))) float v2f;
typedef __attribute__((ext_vector_type(8))) float v8f;

#define NROWS 8192
#define NCOLS 2048

// ---------------- Stage 1: per-block column partial sums ----------------
// Block b accumulates sum over rows {b, b+B1, ...} of (src - tgt) for all
// 2048 columns. Thread t owns cols [4t,4t+4) and [1024+4t,1024+4t+4):
// every b128 load is fully contiguous across the wave (16B lane stride).
__global__ __launch_bounds__(256) void mmd_stage1_colsum(
    const float* __restrict__ src, const float* __restrict__ tgt,
    float* __restrict__ partial, int nBlocks) {
  const int t  = threadIdx.x;      // 0..255
  const int b  = blockIdx.x;       // 0..nBlocks-1
  const int c0 = 4 * t;            // [0, 1024)
  const int c1 = 1024 + 4 * t;     // [1024, 2048)

  float a0x = 0.f, a0y = 0.f, a0z = 0.f, a0w = 0.f;
  float a1x = 0.f, a1y = 0.f, a1z = 0.f, a1w = 0.f;

#pragma unroll 2
  for (int r = b; r < NROWS; r += nBlocks) {
    const size_t base = (size_t)r * NCOLS;
    const float4 s0 = *(const float4*)(src + base + c0);
    const float4 s1 = *(const float4*)(src + base + c1);
    const float4 g0 = *(const float4*)(tgt + base + c0);
    const float4 g1 = *(const float4*)(tgt + base + c1);
    a0x += s0.x - g0.x; a0y += s0.y - g0.y; a0z += s0.z - g0.z; a0w += s0.w - g0.w;
    a1x += s1.x - g1.x; a1y += s1.y - g1.y; a1z += s1.z - g1.z; a1w += s1.w - g1.w;
  }

  float4 o0 = make_float4(a0x, a0y, a0z, a0w);
  float4 o1 = make_float4(a1x, a1y, a1z, a1w);
  float* row = partial + (size_t)b * NCOLS;
  *(float4*)(row + c0) = o0;
  *(float4*)(row + c1) = o1;
}

// ---------------- Stage 2: WMMA reduction across block partials --------
// s[col] = sum_b partial[b][col], computed as ones(16x4) x P-tile chunks on
// the matrix pipe; every row of the f32 accumulator holds the column sums.
// Then sq = s^2 per column, block-reduced. Lanes 16..31 duplicate lanes
// 0..15's columns (C layout: N = lane%16), hence the 0.5 factor.
__global__ __launch_bounds__(128) void mmd_stage2_wmma(
    const float* __restrict__ partial, float* __restrict__ blocksums, int B1) {
  const int lane    = threadIdx.x & 31;
  const int wave    = threadIdx.x >> 5;                 // 0..3
  const int colbase = (blockIdx.x * 4 + wave) * 16;     // covers [0,2048)
  const int n       = lane & 15;
  const int half    = lane >> 4;                        // 0 or 1

  v2f a; a.x = 1.0f; a.y = 1.0f;   // all-ones A (16x4 f32) — column summer
  v8f c = {};                      // 16x16 f32 accumulator

  for (int kk = 0; kk < B1; kk += 4) {
    // Feed 4 partial-rows x 16 cols as the B (4x16 f32) operand. The k-slot
    // assignment across {lane-half, VGPR} is symmetric under ones-A.
    v2f bv;
    bv.x = partial[(size_t)(kk + half * 2)     * NCOLS + colbase + n];
    bv.y = partial[(size_t)(kk + half * 2 + 1) * NCOLS + colbase + n];
    c = __builtin_amdgcn_wmma_f32_16x16x4_f32(
        /*neg_a=*/false, a, /*neg_b=*/false, bv,
        /*c_mod=*/(short)0, c, /*reuse_a=*/false, /*reuse_b=*/false);
  }

  const float s  = c[0];              // D[{0|8}, n] == s[colbase+n]
  const float sq = s * s * 0.5f;      // lanes 16..31 duplicate columns

  __shared__ float red[128];
  red[threadIdx.x] = sq;
  __syncthreads();
  for (int off = 64; off > 0; off >>= 1) {
    if (threadIdx.x < off) red[threadIdx.x] += red[threadIdx.x + off];
    __syncthreads();
  }
  if (threadIdx.x == 0) blocksums[blockIdx.x] = red[0];
}

// ---------------- Stage 3: final reduce + scale ----------------
__global__ __launch_bounds__(32) void mmd_stage3_final(
    const float* __restrict__ blocksums, float* __restrict__ out, int nb) {
  __shared__ float red[32];
  const int t = threadIdx.x;
  red[t] = (t < nb) ? blocksums[t] : 0.0f;
  __syncthreads();
  for (int off = 16; off > 0; off >>= 1) {
    if (t < off) red[t] += red[t + off];
    __syncthreads();
  }
  if (t == 0) {
    // mean over N*N Gram entries; 1/8192^2 is an exact power of two.
    out[0] = red[0] * (1.0f / ((float)NROWS * (float)NROWS));
  }
}

extern "C" void kernel_launch(void* const* d_in, const int* in_sizes, int n_in,
                              void* d_out, int out_size, void* d_ws, size_t ws_size,
                              hipStream_t stream) {
  const float* src = (const float*)d_in[0];   // fea_source [8192, 2048] f32
  const float* tgt = (const float*)d_in[1];   // fea_target [8192, 2048] f32
  float* out = (float*)d_out;                 // scalar f32
  float* partial = (float*)d_ws;

  // Stage-1 block count: want 256 (2 MB of partials); shrink (multiple of 4)
  // if scratch is smaller. Deterministic given fixed ws_size.
  int B1 = 256;
  const size_t perBlock = (size_t)NCOLS * sizeof(float);
  if (ws_size < (size_t)B1 * perBlock + 256) {
    size_t m = (ws_size > 256) ? (ws_size - 256) / perBlock : 4;
    B1 = (int)m;
    B1 &= ~3;
    if (B1 < 4) B1 = 4;
    if (B1 > 256) B1 = 256;
  }
  float* blocksums = partial + (size_t)B1 * NCOLS;   // 32 floats

  hipLaunchKernelGGL(mmd_stage1_colsum, dim3(B1), dim3(256), 0, stream,
                     src, tgt, partial, B1);
  hipLaunchKernelGGL(mmd_stage2_wmma, dim3(32), dim3(128), 0, stream,
                     partial, blocksums, B1);
  hipLaunchKernelGGL(mmd_stage3_final, dim3(1), dim3(32), 0, stream,
                     blocksums, out, 32);
}